// CircularShift_21844203667967
// MI455X (gfx1250) — compile-verified
//
#include <hip/hip_runtime.h>
#include <stdint.h>

// out[r, j] = in[r, (j-1) mod D]  — circular right-shift of each row by 1.
// Pure data movement: 256 MiB total traffic -> ~11.5 us at 23.3 TB/s HBM.
// Strategy: async DMA (global -> LDS, CDNA5 ASYNCcnt path) with fully aligned
// b128 traffic on both global sides; the one-element shift is absorbed by
// dword-granularity LDS reads (LDS b32 has no alignment hazard).

#define D_COLS 4096          // row length (floats), known from the reference
#define BLOCK  256           // 8 waves (wave32)

__global__ __launch_bounds__(BLOCK) void circshift_row_kernel(
    const float* __restrict__ in, float* __restrict__ out)
{
    __shared__ __align__(16) float smem[D_COLS];

    const int    tid = threadIdx.x;
    const size_t row = (size_t)blockIdx.x;
    const float* rin  = in  + row * D_COLS;
    float*       rout = out + row * D_COLS;

    // -------- Phase 1: async copy row -> LDS (aligned b128 both sides) ----
    // Each float4 chunk ch: global [rin + 4*ch .. +3] -> smem[4*ch .. +3].
    // Consecutive lanes take consecutive chunks => coalesced 512B/wave/issue.
#pragma unroll
    for (int k = 0; k < D_COLS / (BLOCK * 4); ++k) {
        const int ch = k * BLOCK + tid;                 // float4 chunk index
        const float* g = rin + 4 * ch;                  // 16B-aligned global
        const uint32_t l =
            (uint32_t)(uintptr_t)(const void*)(&smem[4 * ch]);  // LDS byte addr
        asm volatile("global_load_async_to_lds_b128 %0, %1, off"
                     :: "v"(l), "v"(g)
                     : "memory");
    }

    // Wait for this wave's async transfers (ASYNCcnt), then make the LDS
    // contents visible across all 8 waves of the workgroup.
    asm volatile("s_wait_asynccnt 0" ::: "memory");
    __syncthreads();

    // -------- Phase 2: shifted LDS read, aligned b128 global store --------
    // out[j0+0..3] = in[(j0-1)&4095], in[j0], in[j0+1], in[j0+2]
#pragma unroll
    for (int k = 0; k < D_COLS / (BLOCK * 4); ++k) {
        const int ch = k * BLOCK + tid;
        const int j0 = 4 * ch;
        float4 v;
        v.x = smem[(j0 - 1) & (D_COLS - 1)];   // wrap handled by mask
        v.y = smem[j0 + 0];
        v.z = smem[j0 + 1];
        v.w = smem[j0 + 2];
        *reinterpret_cast<float4*>(rout + j0) = v;   // 16B-aligned store
    }
}

extern "C" void kernel_launch(void* const* d_in, const int* in_sizes, int n_in,
                              void* d_out, int out_size, void* d_ws, size_t ws_size,
                              hipStream_t stream) {
    const float* in  = (const float*)d_in[0];
    float*       out = (float*)d_out;
    const int rows = in_sizes[0] / D_COLS;   // 8192 for the reference shapes
    circshift_row_kernel<<<dim3(rows), dim3(BLOCK), 0, stream>>>(in, out);
}